// GNAT_block_14388140442035
// MI455X (gfx1250) — compile-verified
//
#include <hip/hip_runtime.h>
#include <hip/hip_bf16.h>

typedef __attribute__((ext_vector_type(2))) float v2f;
typedef __attribute__((ext_vector_type(8))) float v8f;

// ---------------------------------------------------------------------------
// WMMA helpers: D = A(16x4,f32) * B(4x16,f32) + C(16x16,f32)
// A frag: lane l holds A[l&15][k0 + 2*(l>>4) + {0,1}]
// B frag: lane l holds B[k0 + 2*(l>>4) + {0,1}][n0 + (l&15)]  (B = W^T)
// C frag: VGPR i, lane l -> (m = i + 8*(l>>4), n = n0 + (l&15))
// ---------------------------------------------------------------------------
__device__ __forceinline__ v8f wmma_f32(v2f a, v2f b, v8f c) {
  return __builtin_amdgcn_wmma_f32_16x16x4_f32(false, a, false, b, (short)0, c,
                                               false, false);
}

__device__ __forceinline__ v2f lds_a_frag(const float* A, int lda, int k0, int lane) {
  const int m = lane & 15;
  const int k = k0 + ((lane >> 4) << 1);
  v2f r;
  r.x = A[m * lda + k];
  r.y = A[m * lda + k + 1];
  return r;
}

// W is row-major [OUT][IN]; B[k][n] = W[n][k]
__device__ __forceinline__ v2f w_b_frag(const float* W, int ldw, int n0, int k0, int lane) {
  const int n = n0 + (lane & 15);
  const int k = k0 + ((lane >> 4) << 1);
  v2f r;
  r.x = W[(size_t)n * ldw + k];
  r.y = W[(size_t)n * ldw + k + 1];
  return r;
}

__device__ __forceinline__ void atomicMaxF32(float* addr, float val) {
  unsigned int* ua = (unsigned int*)addr;
  unsigned int old = *ua;
  while (true) {
    float f = __uint_as_float(old);
    if (f >= val) break;
    unsigned int assumed = old;
    old = atomicCAS(ua, assumed, __float_as_uint(val));
    if (old == assumed) break;
  }
}

// CDNA5 async global->LDS copy (per-lane addresses), tracked by ASYNCcnt.
__device__ __forceinline__ void async_copy_b128(unsigned lds_addr,
                                                const void* gaddr) {
  asm volatile("global_load_async_to_lds_b128 %0, %1, off"
               :
               : "v"(lds_addr), "v"(gaddr)
               : "memory");
}
__device__ __forceinline__ void wait_async0() {
  asm volatile("s_wait_asynccnt 0x0" ::: "memory");
}

// ---------------------------------------------------------------------------
// init workspace
// ---------------------------------------------------------------------------
__global__ void init_kernel(float* __restrict__ segmax, float* __restrict__ denom,
                            float* __restrict__ agg, float* __restrict__ gsum,
                            float* __restrict__ cnt, int N, int G) {
  const int i = blockIdx.x * blockDim.x + threadIdx.x;
  if (i < N * 128) agg[i] = 0.f;
  if (i < N) {
    segmax[i] = -__builtin_inff();
    denom[i] = 0.f;
  }
  if (i < G * 128) gsum[i] = 0.f;
  if (i < G) cnt[i] = 0.f;
}

__global__ void cnt_kernel(const int* __restrict__ batch, float* __restrict__ cnt, int N) {
  const int i = blockIdx.x * blockDim.x + threadIdx.x;
  if (i < N) atomicAdd(&cnt[batch[i]], 1.0f);
}

// ---------------------------------------------------------------------------
// feat = x@W_node^T + b ; key = x@W_key^T + b   (16-node tiles, 4 waves)
// ---------------------------------------------------------------------------
__global__ __launch_bounds__(128) void node_lin_kernel(
    const float* __restrict__ x, const float* __restrict__ W_node,
    const float* __restrict__ b_node, const float* __restrict__ W_key,
    const float* __restrict__ b_key, float* __restrict__ feat,
    float* __restrict__ key, int N) {
  __shared__ __align__(16) float xs[16 * 128];
  const int tid = threadIdx.x;
  const int m0 = blockIdx.x * 16;
  {
    const int row = tid >> 3, sub = tid & 7;  // 8 threads per row, 4 float4 each
    const int node = min(m0 + row, N - 1);
    const float* src = x + (size_t)node * 128;
    const unsigned lds_base = (unsigned)(size_t)(xs + row * 128);
#pragma unroll
    for (int j = 0; j < 4; ++j) {
      const int idx = sub + 8 * j;  // float4 index 0..31
      async_copy_b128(lds_base + idx * 16, src + idx * 4);
    }
    wait_async0();
  }
  __syncthreads();
  const int lane = tid & 31, wave = tid >> 5;
  const int half = lane >> 4, nl = lane & 15;
#pragma unroll
  for (int which = 0; which < 2; ++which) {
    const float* W = which ? W_key : W_node;
    const float* B = which ? b_key : b_node;
    float* O = which ? key : feat;
#pragma unroll
    for (int t = 0; t < 2; ++t) {
      const int n0 = (wave * 2 + t) * 16;
      v8f acc = {};
      for (int k0 = 0; k0 < 128; k0 += 4) {
        v2f a = lds_a_frag(xs, 128, k0, lane);
        v2f b = w_b_frag(W, 128, n0, k0, lane);
        acc = wmma_f32(a, b, acc);
      }
      const int n = n0 + nl;
      const float bv = B[n];
#pragma unroll
      for (int i = 0; i < 8; ++i) {
        const int m = i + 8 * half;
        if (m0 + m < N) O[(size_t)(m0 + m) * 128 + n] = acc[i] + bv;
      }
    }
  }
}

// ---------------------------------------------------------------------------
// Fused edge kernel: e = relu([x[S]||ea] @ W_edge^T + b_edge)  (store to out)
//                    code = e @ W_code^T + b_code
//                    logits[e] = <key[R], code>, atomicMax segmax[R]
// Staging of the gathered 16x256 concat tile uses CDNA5 async global->LDS.
// ---------------------------------------------------------------------------
__global__ __launch_bounds__(128) void edge_kernel(
    const float* __restrict__ x, const int* __restrict__ ei,
    const float* __restrict__ eattr, const float* __restrict__ W_edge,
    const float* __restrict__ b_edge, const float* __restrict__ W_code,
    const float* __restrict__ b_code, const float* __restrict__ key,
    float* __restrict__ e_out, float* __restrict__ logits,
    float* __restrict__ segmax, int E) {
  __shared__ __align__(16) float xs[16 * 256];
  __shared__ __align__(16) float et[16 * 128];
  __shared__ __align__(16) float ct[16 * 128];
  __shared__ float lp[16 * 8];
  __shared__ int Sg[16], Rg[16];
  const int tid = threadIdx.x;
  const int e0 = blockIdx.x * 16;
  if (tid < 16) {
    const int ee = min(e0 + tid, E - 1);
    Sg[tid] = ei[ee];
    Rg[tid] = ei[E + ee];
  }
  __syncthreads();
  {  // stage concat rows 16 x 256 directly into LDS (async, per-lane gather)
    const int row = tid >> 3, sub = tid & 7;
    const int ee = min(e0 + row, E - 1);
    const float* sx = x + (size_t)Sg[row] * 128;
    const float* se = eattr + (size_t)ee * 128;
    const unsigned lds_base = (unsigned)(size_t)(xs + row * 256);
#pragma unroll
    for (int j = 0; j < 8; ++j) {
      const int idx = sub + 8 * j;  // float4 index 0..63 within row
      const float* g = (idx < 32) ? (sx + idx * 4) : (se + (idx - 32) * 4);
      async_copy_b128(lds_base + idx * 16, g);
    }
    wait_async0();
  }
  __syncthreads();
  const int lane = tid & 31, wave = tid >> 5;
  const int half = lane >> 4, nl = lane & 15;
  // ---- e tiles: 8 n-tiles over K=256, 2 per wave ----
#pragma unroll
  for (int t = 0; t < 2; ++t) {
    const int n0 = (wave * 2 + t) * 16;
    v8f acc = {};
    for (int k0 = 0; k0 < 256; k0 += 4) {
      v2f a = lds_a_frag(xs, 256, k0, lane);
      v2f b = w_b_frag(W_edge, 256, n0, k0, lane);
      acc = wmma_f32(a, b, acc);
    }
    const int n = n0 + nl;
    const float bv = b_edge[n];
#pragma unroll
    for (int i = 0; i < 8; ++i) {
      const int m = i + 8 * half;
      const float v = fmaxf(acc[i] + bv, 0.f);
      et[m * 128 + n] = v;
      if (e0 + m < E) e_out[(size_t)(e0 + m) * 128 + n] = v;
    }
  }
  __syncthreads();
  // ---- code tiles: 8 n-tiles over K=128 from LDS e tile ----
#pragma unroll
  for (int t = 0; t < 2; ++t) {
    const int n0 = (wave * 2 + t) * 16;
    v8f acc = {};
    for (int k0 = 0; k0 < 128; k0 += 4) {
      v2f a = lds_a_frag(et, 128, k0, lane);
      v2f b = w_b_frag(W_code, 128, n0, k0, lane);
      acc = wmma_f32(a, b, acc);
    }
    const int n = n0 + nl;
    const float bv = b_code[n];
#pragma unroll
    for (int i = 0; i < 8; ++i) {
      const int m = i + 8 * half;
      ct[m * 128 + n] = acc[i] + bv;
    }
  }
  __syncthreads();
  // ---- logits: 8 threads per edge, 16-float partials ----
  {
    const int em = tid & 15, chunk = tid >> 4;  // chunk 0..7
    const float* kr = key + (size_t)Rg[em] * 128;
    float part = 0.f;
#pragma unroll
    for (int j = 0; j < 16; ++j)
      part += kr[chunk * 16 + j] * ct[em * 128 + chunk * 16 + j];
    lp[em * 8 + chunk] = part;
  }
  __syncthreads();
  if (tid < 16 && e0 + tid < E) {
    float s = 0.f;
#pragma unroll
    for (int j = 0; j < 8; ++j) s += lp[tid * 8 + j];
    logits[e0 + tid] = s;
    atomicMaxF32(&segmax[Rg[tid]], s);
  }
}

// ---------------------------------------------------------------------------
// ex = exp(logit - segmax[R]); denom[R] += ex
// ---------------------------------------------------------------------------
__global__ void ex_kernel(const float* __restrict__ logits,
                          const int* __restrict__ ei,
                          const float* __restrict__ segmax,
                          float* __restrict__ exv, float* __restrict__ denom,
                          int E) {
  const int e = blockIdx.x * blockDim.x + threadIdx.x;
  if (e >= E) return;
  const int r = ei[E + e];
  float m = segmax[r];
  if (!(m > -1e37f && m < 1e37f)) m = 0.f;
  const float v = __expf(logits[e] - m);
  exv[e] = v;
  atomicAdd(&denom[r], v);
}

// ---------------------------------------------------------------------------
// agg[R] += e * attention   (32 lanes/edge, float4 each)
// ---------------------------------------------------------------------------
__global__ void agg_kernel(const float* __restrict__ e_out,
                           const float* __restrict__ exv,
                           const float* __restrict__ denom,
                           const int* __restrict__ ei, float* __restrict__ agg,
                           int E) {
  const int gid = blockIdx.x * blockDim.x + threadIdx.x;
  const int e = gid >> 5;
  const int f4 = gid & 31;
  if (e >= E) return;
  const int r = ei[E + e];
  const float att = exv[e] / (denom[r] + 1e-16f);
  const float4 v = ((const float4*)(e_out + (size_t)e * 128))[f4];
  float* dst = agg + (size_t)r * 128 + f4 * 4;
  atomicAdd(dst + 0, v.x * att);
  atomicAdd(dst + 1, v.y * att);
  atomicAdd(dst + 2, v.z * att);
  atomicAdd(dst + 3, v.w * att);
}

// ---------------------------------------------------------------------------
// node_out = relu([feat||agg||glob[batch]] @ W_node2^T + b2); gsum += node_out
// ---------------------------------------------------------------------------
__global__ __launch_bounds__(128) void node2_kernel(
    const float* __restrict__ feat, const float* __restrict__ agg,
    const float* __restrict__ glob, const int* __restrict__ batch,
    const float* __restrict__ W2, const float* __restrict__ b2,
    float* __restrict__ node_out, float* __restrict__ gsum, int N) {
  __shared__ __align__(16) float cs[16 * 384];
  __shared__ int bt[16];
  const int tid = threadIdx.x;
  const int m0 = blockIdx.x * 16;
  if (tid < 16) bt[tid] = batch[min(m0 + tid, N - 1)];
  __syncthreads();
  {
    const int row = tid >> 3, sub = tid & 7;  // 8 thr/row, 96 float4/row
    const int node = min(m0 + row, N - 1);
    const float* f4 = feat + (size_t)node * 128;
    const float* a4 = agg + (size_t)node * 128;
    const float* g4 = glob + (size_t)bt[row] * 128;
    const unsigned lds_base = (unsigned)(size_t)(cs + row * 384);
#pragma unroll
    for (int j = 0; j < 12; ++j) {
      const int idx = sub + 8 * j;  // float4 index 0..95
      const float* g = (idx < 32)   ? (f4 + idx * 4)
                       : (idx < 64) ? (a4 + (idx - 32) * 4)
                                    : (g4 + (idx - 64) * 4);
      async_copy_b128(lds_base + idx * 16, g);
    }
    wait_async0();
  }
  __syncthreads();
  const int lane = tid & 31, wave = tid >> 5;
  const int half = lane >> 4, nl = lane & 15;
#pragma unroll
  for (int t = 0; t < 2; ++t) {
    const int n0 = (wave * 2 + t) * 16;
    v8f acc = {};
    for (int k0 = 0; k0 < 384; k0 += 4) {
      v2f a = lds_a_frag(cs, 384, k0, lane);
      v2f b = w_b_frag(W2, 384, n0, k0, lane);
      acc = wmma_f32(a, b, acc);
    }
    const int n = n0 + nl;
    const float bv = b2[n];
#pragma unroll
    for (int i = 0; i < 8; ++i) {
      const int m = i + 8 * half;
      if (m0 + m < N) {
        const float v = fmaxf(acc[i] + bv, 0.f);
        node_out[(size_t)(m0 + m) * 128 + n] = v;
        atomicAdd(&gsum[(size_t)bt[m] * 128 + n], v);
      }
    }
  }
}

// ---------------------------------------------------------------------------
// glob_out = relu([glob||mean_n] @ W_glob^T + bg)  (single block, G=64)
// ---------------------------------------------------------------------------
__global__ __launch_bounds__(256) void glob_kernel(
    const float* __restrict__ glob, const float* __restrict__ gsum,
    const float* __restrict__ cnt, const float* __restrict__ Wg,
    const float* __restrict__ bg, float* __restrict__ glob_out, int G) {
  __shared__ __align__(16) float cs[64 * 256];
  const int tid = threadIdx.x;
#pragma unroll
  for (int j = 0; j < 64; ++j) {
    const int idx = tid + 256 * j;  // 0..16383
    const int row = idx >> 8;
    const int col = idx & 255;
    const int g = min(row, G - 1);
    float v;
    if (col < 128)
      v = glob[(size_t)g * 128 + col];
    else
      v = gsum[(size_t)g * 128 + (col - 128)] / fmaxf(cnt[g], 1.0f);
    cs[idx] = v;
  }
  __syncthreads();
  const int lane = tid & 31, wave = tid >> 5;
  const int half = lane >> 4, nl = lane & 15;
#pragma unroll
  for (int t = 0; t < 4; ++t) {  // 8 waves x 4 tiles = 32 (16x16) tiles
    const int tt = wave * 4 + t;
    const int mt = tt >> 3, nt = tt & 7;
    const int n0 = nt * 16;
    v8f acc = {};
    for (int k0 = 0; k0 < 256; k0 += 4) {
      v2f a = lds_a_frag(cs + mt * 16 * 256, 256, k0, lane);
      v2f b = w_b_frag(Wg, 256, n0, k0, lane);
      acc = wmma_f32(a, b, acc);
    }
    const int n = n0 + nl;
    const float bv = bg[n];
#pragma unroll
    for (int i = 0; i < 8; ++i) {
      const int m = mt * 16 + i + 8 * half;
      if (m < G) glob_out[(size_t)m * 128 + n] = fmaxf(acc[i] + bv, 0.f);
    }
  }
}

// ---------------------------------------------------------------------------
extern "C" void kernel_launch(void* const* d_in, const int* in_sizes, int n_in,
                              void* d_out, int out_size, void* d_ws,
                              size_t ws_size, hipStream_t stream) {
  (void)n_in;
  (void)out_size;
  (void)ws_size;
  const float* x = (const float*)d_in[0];
  const int* ei = (const int*)d_in[1];
  const float* eattr = (const float*)d_in[2];
  const float* glob = (const float*)d_in[3];
  const int* batch = (const int*)d_in[4];
  const float* W_edge = (const float*)d_in[5];
  const float* b_edge = (const float*)d_in[6];
  const float* W_code = (const float*)d_in[7];
  const float* b_code = (const float*)d_in[8];
  const float* W_key = (const float*)d_in[9];
  const float* b_key = (const float*)d_in[10];
  const float* W_node = (const float*)d_in[11];
  const float* b_node = (const float*)d_in[12];
  const float* W_node2 = (const float*)d_in[13];
  const float* b_node2 = (const float*)d_in[14];
  const float* W_glob = (const float*)d_in[15];
  const float* b_glob = (const float*)d_in[16];

  const int N = in_sizes[0] / 128;
  const int E = in_sizes[2] / 128;
  const int G = in_sizes[3] / 128;
  const size_t ND = (size_t)N * 128;
  const size_t ED = (size_t)E * 128;

  float* node_out = (float*)d_out;
  float* e_out = node_out + ND;
  float* glob_out = e_out + ED;

  float* ws = (float*)d_ws;
  float* feat = ws;
  float* key = ws + ND;
  float* agg = ws + 2 * ND;
  float* segmax = ws + 3 * ND;
  float* denom = segmax + N;
  float* logits = denom + N;
  float* exv = logits + E;
  float* gsum = exv + E;
  float* cnt = gsum + (size_t)G * 128;

  init_kernel<<<(N * 128 + 255) / 256, 256, 0, stream>>>(segmax, denom, agg,
                                                         gsum, cnt, N, G);
  cnt_kernel<<<(N + 255) / 256, 256, 0, stream>>>(batch, cnt, N);
  node_lin_kernel<<<(N + 15) / 16, 128, 0, stream>>>(x, W_node, b_node, W_key,
                                                     b_key, feat, key, N);
  edge_kernel<<<(E + 15) / 16, 128, 0, stream>>>(x, ei, eattr, W_edge, b_edge,
                                                 W_code, b_code, key, e_out,
                                                 logits, segmax, E);
  ex_kernel<<<(E + 255) / 256, 256, 0, stream>>>(logits, ei, segmax, exv,
                                                 denom, E);
  {
    const long long tot = (long long)E * 32;
    agg_kernel<<<(unsigned int)((tot + 255) / 256), 256, 0, stream>>>(
        e_out, exv, denom, ei, agg, E);
  }
  node2_kernel<<<(N + 15) / 16, 128, 0, stream>>>(
      feat, agg, glob, batch, W_node2, b_node2, node_out, gsum, N);
  glob_kernel<<<1, 256, 0, stream>>>(glob, gsum, cnt, W_glob, b_glob, glob_out,
                                     G);
}